// Decoder_74947179315482
// MI455X (gfx1250) — compile-verified
//
#include <hip/hip_runtime.h>
#include <hip/hip_bf16.h>
#include <math.h>

// Problem dims (must match the reference)
#define BB   64
#define SS   1024
#define TT   400
#define VV   1000
#define HH   512
#define EE   256
#define KVD  128

#define NWG          64
#define NTHREADS     256
#define WAVES_PER_WG 8

// LDS panels (float offsets). All strides are 4 (mod 64) -> conflict-free b64 reads.
//  A_EMB : 64 rows x 260  (256-wide rows, TDM pad 4dw/256dw)      -> emb or ctx
//  A_CTX : 64 rows x 132  (128-wide rows, TDM pad 4dw/128dw)      -> ctx or h2
//  A_H1  : 64 rows x 516  (512-wide rows, TDM pad 2dw/256dw; mid-row pad at col 256)
#define A_EMB 0
#define A_CTX (64 * 260)                 // 16640
#define A_H1  (A_CTX + 64 * 132)         // 25088
#define SMEM_FLOATS (A_H1 + 64 * 516)    // 58112 floats = 232448 B  (< 320KB WGP LDS)

typedef __attribute__((ext_vector_type(2))) float v2f;
typedef __attribute__((ext_vector_type(8))) float v8f;
typedef __attribute__((ext_vector_type(4))) unsigned int v4u;
typedef __attribute__((ext_vector_type(4))) int v4i;
typedef __attribute__((ext_vector_type(8))) int v8i;

__device__ __forceinline__ float sigmf(float x) { return 1.0f / (1.0f + __expf(-x)); }

// ======================= Tensor Data Mover (gfx1250) =======================
#if __has_builtin(__builtin_amdgcn_tensor_load_to_lds)
#define HAVE_TDM 1
#if defined(__clang_major__) && (__clang_major__ >= 23)
#define TDM_LOAD(g0, g1, g2, g3) \
  __builtin_amdgcn_tensor_load_to_lds((g0), (g1), (g2), (g3), (v8i)0, 0)
#else
#define TDM_LOAD(g0, g1, g2, g3) \
  __builtin_amdgcn_tensor_load_to_lds((g0), (g1), (g2), (g3), 0)
#endif

// Contiguous 1-D global->LDS DMA of `ndw` dwords with LDS padding:
// pad (pa+1) dwords inserted after every 2^pi 8-byte units (per D# spec).
__device__ __forceinline__ void tdm_load_contig(const float* gsrc, unsigned lds_byte,
                                                unsigned ndw, unsigned pi, unsigned pa) {
  unsigned long long ga = (unsigned long long)gsrc;
  v4u g0;
  g0[0] = 1u;                                        // count=1, no gather
  g0[1] = lds_byte;                                  // lds_addr (bytes)
  g0[2] = (unsigned)ga;                              // global_addr[31:0]
  g0[3] = (unsigned)((ga >> 32) & 0x1FFFFFFull) | (2u << 30);  // [56:32] | type=2
  v8i g1;
  g1[0] = (int)((2u << 16) | (1u << 20) | (pi << 22) | (pa << 25)); // 4B elts, pad_enable
  g1[1] = (int)((ndw & 0xffffu) << 16);              // tensor_dim0 lo16
  g1[2] = (int)(((ndw >> 16) & 0xffffu) | (1u << 16)); // tensor_dim0 hi16 | tensor_dim1=1
  g1[3] = (int)((ndw & 0xffffu) << 16);              // tile_dim0 = ndw (<= 65535)
  g1[4] = 0;                                         // tile_dim1/2 unused
  g1[5] = (int)ndw;                                  // tensor_dim0_stride lo32
  g1[6] = 0; g1[7] = 0;
  v4i z4 = (v4i)0;
  TDM_LOAD(g0, g1, z4, z4);
}

// Gather-mode DMA: 16 rows of W_emb (row width 256 dwords), row ids in g2/g3
// (16-bit indices), LDS rows padded to stride 260.
__device__ __forceinline__ void tdm_gather16(const float* Wemb, unsigned lds_byte,
                                             v4i g2, v4i g3) {
  unsigned long long ga = (unsigned long long)Wemb;
  v4u g0;
  g0[0] = 1u | (1u << 31);                           // count=1 | gather_mode (16-bit idx)
  g0[1] = lds_byte;
  g0[2] = (unsigned)ga;
  g0[3] = (unsigned)((ga >> 32) & 0x1FFFFFFull) | (2u << 30);
  v8i g1;
  g1[0] = (int)((2u << 16) | (1u << 20) | (7u << 22) | (3u << 25)); // pad 4dw / 256dw
  g1[1] = (int)((256u & 0xffffu) << 16);             // tensor_dim0 = 256
  g1[2] = (int)((1000u & 0xffffu) << 16);            // tensor_dim1 = 1000 rows
  g1[3] = (int)(256u << 16);                         // tile_dim0 = 256
  g1[4] = 16;                                        // tile_dim1 = #indices
  g1[5] = 256;                                       // tensor_dim0_stride = 256
  g1[6] = 0; g1[7] = 0;
  TDM_LOAD(g0, g1, g2, g3);
}
#else
#define HAVE_TDM 0
#endif

// ================= grid-wide sense barrier (device scope) =================
__device__ __forceinline__ void grid_barrier(unsigned* cnt, unsigned* gen) {
  __syncthreads();
  if (threadIdx.x == 0) {
    unsigned g = __hip_atomic_load(gen, __ATOMIC_ACQUIRE, __HIP_MEMORY_SCOPE_AGENT);
    __threadfence();
    if (__hip_atomic_fetch_add(cnt, 1u, __ATOMIC_ACQ_REL, __HIP_MEMORY_SCOPE_AGENT) ==
        (unsigned)(NWG - 1)) {
      __hip_atomic_store(cnt, 0u, __ATOMIC_RELAXED, __HIP_MEMORY_SCOPE_AGENT);
      __hip_atomic_fetch_add(gen, 1u, __ATOMIC_RELEASE, __HIP_MEMORY_SCOPE_AGENT);
    } else {
      while (__hip_atomic_load(gen, __ATOMIC_ACQUIRE, __HIP_MEMORY_SCOPE_AGENT) == g)
        __builtin_amdgcn_s_sleep(2);
    }
  }
  __syncthreads();
}

// ================= fp32 WMMA span: C += A(LDS row) * W(row)^T =================
// As: this lane's A row segment base in LDS; Wrow: this lane's weight row segment.
// A layout: lanes 0-15 K=k,k+1 ; lanes 16-31 K=k+2,k+3 (v_wmma_f32_16x16x4_f32).
__device__ __forceinline__ v8f wmma_span(v8f c, const float* __restrict__ As,
                                         int len, const float* __restrict__ Wrow) {
  const int hi2 = (threadIdx.x >> 4) & 1;
  #pragma unroll 4
  for (int k = 0; k < len; k += 4) {
    v2f a = *(const v2f*)(As + k + 2 * hi2);
    v2f b = *(const v2f*)(Wrow + k + 2 * hi2);
    c = __builtin_amdgcn_wmma_f32_16x16x4_f32(false, a, false, b, (short)0, c,
                                              false, false);
  }
  return c;
}

// C/D layout: lanes 0-15 -> N=lane, M=v ; lanes 16-31 -> N=lane-16, M=v+8.
__device__ __forceinline__ void store_tile(v8f c, float* __restrict__ out, long ldo,
                                           int m0, int n, int Nmax) {
  const int hi2 = (threadIdx.x >> 4) & 1;
  if (n < Nmax) {
    #pragma unroll
    for (int v = 0; v < 8; ++v)
      out[(long)(m0 + v + 8 * hi2) * ldo + n] = c[v];
  }
}

// ================= masked softmax attention: one batch per WG =================
__device__ __forceinline__ void attention_phase(
    float* smem, const float* __restrict__ h2q,
    const float* __restrict__ key, const float* __restrict__ val,
    const int* __restrict__ elen, float* __restrict__ ctx,
    float* __restrict__ attn_out)
{
  const int b   = blockIdx.x;     // NWG == BB
  const int tid = threadIdx.x;
  float* sE  = smem;              // [1024]
  float* q   = smem + SS;         // [128]
  float* red = smem + SS + KVD;   // [256]

  __syncthreads();
  for (int d = tid; d < KVD; d += NTHREADS) q[d] = h2q ? h2q[b * KVD + d] : 0.0f;
  __syncthreads();

  const int len = elen[b];
  const float scale = 0.08838834764831845f;   // 1/sqrt(128)

  float lmax = -INFINITY;
  for (int s = tid; s < SS; s += NTHREADS) {
    float e = -INFINITY;
    if (s < len) {
      const float* kr = key + ((long)b * SS + s) * KVD;
      float acc = 0.0f;
      #pragma unroll 8
      for (int d = 0; d < KVD; ++d) acc = fmaf(q[d], kr[d], acc);
      e = acc * scale;
    }
    sE[s] = e;
    lmax = fmaxf(lmax, e);
  }
  red[tid] = lmax; __syncthreads();
  for (int off = NTHREADS / 2; off > 0; off >>= 1) {
    if (tid < off) red[tid] = fmaxf(red[tid], red[tid + off]);
    __syncthreads();
  }
  const float m = red[0];
  __syncthreads();

  float lsum = 0.0f;
  for (int s = tid; s < SS; s += NTHREADS) {
    float w = (s < len) ? __expf(sE[s] - m) : 0.0f;
    sE[s] = w;
    lsum += w;
  }
  red[tid] = lsum; __syncthreads();
  for (int off = NTHREADS / 2; off > 0; off >>= 1) {
    if (tid < off) red[tid] += red[tid + off];
    __syncthreads();
  }
  const float inv = 1.0f / red[0];
  __syncthreads();

  const int d    = tid & (KVD - 1);
  const int half = tid >> 7;
  float acc = 0.0f;
  for (int s = half * (SS / 2); s < (half + 1) * (SS / 2); ++s)
    acc = fmaf(sE[s], val[((long)b * SS + s) * KVD + d], acc);
  red[tid] = acc; __syncthreads();
  if (tid < KVD) ctx[b * KVD + tid] = (red[tid] + red[tid + KVD]) * inv;

  if (attn_out && b == 0)
    for (int s = tid; s < SS; s += NTHREADS) attn_out[s] = sE[s] * inv;
  __syncthreads();
}

// ================= init: zero recurrent state + barrier =================
__global__ void las_init(float* ws) {
  float* h1 = ws + (long)BB * 4 * HH + (long)BB * 4 * KVD;   // h1,c1,h2,c2 contiguous
  const long nstate = 2L * BB * HH + 2L * BB * KVD;
  const long gsz = (long)gridDim.x * blockDim.x;
  for (long i = (long)blockIdx.x * blockDim.x + threadIdx.x; i < nstate; i += gsz)
    h1[i] = 0.0f;
  if (blockIdx.x == 0 && threadIdx.x == 0) {
    unsigned* bar = (unsigned*)(ws + (long)BB * 4 * HH + (long)BB * 4 * KVD +
                                2L * BB * HH + 3L * BB * KVD);
    bar[0] = 0u; bar[1] = 0u;
  }
}

// ================= persistent decoder: T=400 scan in one launch =================
__global__ void __launch_bounds__(NTHREADS) las_decoder(
    const float* __restrict__ enc_key, const float* __restrict__ enc_value,
    const int*   __restrict__ enc_len, const int* __restrict__ y,
    const float* __restrict__ W_emb,  const float* __restrict__ b_out,
    const float* __restrict__ W_ih1,  const float* __restrict__ W_hh1,
    const float* __restrict__ b_ih1,  const float* __restrict__ b_hh1,
    const float* __restrict__ W_ih2,  const float* __restrict__ W_hh2,
    const float* __restrict__ b_ih2,  const float* __restrict__ b_hh2,
    float* __restrict__ d_out, float* __restrict__ ws)
{
  extern __shared__ float smem[];

  // global scratch (L2-resident, ~1 MB)
  float* gates1 = ws;                         // [64 x 2048]
  float* gates2 = gates1 + (long)BB * 4 * HH; // [64 x 512]
  float* h1  = gates2 + (long)BB * 4 * KVD;   // [64 x 512]
  float* c1  = h1 + (long)BB * HH;
  float* h2  = c1 + (long)BB * HH;            // [64 x 128]
  float* c2  = h2 + (long)BB * KVD;
  float* ctx = c2 + (long)BB * KVD;
  unsigned* bar = (unsigned*)(ctx + (long)BB * KVD);

  float* preds = d_out;                       // [B,T,V]
  float* attns = d_out + (long)BB * TT * VV;  // [T,S]

  const int tid  = threadIdx.x;
  const int wave = tid >> 5;
  const int tix  = blockIdx.x * WAVES_PER_WG + wave;   // 0..511
  const int ln   = tid & 15;

  // initial context: attention with zero query (uniform over valid positions)
  attention_phase(smem, nullptr, enc_key, enc_value, enc_len, ctx, nullptr);
  grid_barrier(bar, bar + 1);

  for (int t = 0; t < TT; ++t) {
    // ---- Phase 1 stage: emb gather -> A_EMB, ctx -> A_CTX, h1 -> A_H1 ----
#if HAVE_TDM
    if (wave == 0) {
      tdm_load_contig(ctx, (unsigned)(A_CTX * 4), BB * KVD, 6u, 3u);  // stride-132 rows
      tdm_load_contig(h1,  (unsigned)(A_H1 * 4),  BB * HH,  7u, 1u);  // stride-516 rows
    }
    if (wave < 4) {
      const int r = wave * 16 + ln;
      int tok = (t == 0) ? 0 : y[r * TT + (t - 1)];
      int pk[8];
      #pragma unroll
      for (int i = 0; i < 8; ++i) {
        int lo = __builtin_amdgcn_readlane(tok, 2 * i);
        int hh = __builtin_amdgcn_readlane(tok, 2 * i + 1);
        pk[i] = (lo & 0xffff) | (hh << 16);
      }
      v4i g2 = {pk[0], pk[1], pk[2], pk[3]};
      v4i g3 = {pk[4], pk[5], pk[6], pk[7]};
      tdm_gather16(W_emb, (unsigned)((A_EMB + wave * 16 * 260) * 4), g2, g3);
      __builtin_amdgcn_s_wait_tensorcnt(0);
    }
#else
    for (int idx = tid; idx < BB * EE; idx += NTHREADS) {
      const int r = idx >> 8, k = idx & 255;
      const int tok = (t == 0) ? 0 : y[r * TT + (t - 1)];
      smem[A_EMB + r * 260 + k] = W_emb[(long)tok * EE + k];
    }
    for (int idx = tid; idx < BB * KVD; idx += NTHREADS) {
      const int r = idx >> 7, k = idx & 127;
      smem[A_CTX + r * 132 + k] = ctx[idx];
    }
    for (int idx = tid; idx < BB * HH; idx += NTHREADS) {
      const int r = idx >> 9, k = idx & 511;
      smem[A_H1 + r * 516 + k + 2 * (k >= 256)] = h1[idx];
    }
#endif
    __syncthreads();

    // ---- Phase 1 WMMA: gates1 = [emb|ctx|h1] @ [W_ih1|W_hh1]^T + b ----
    {
      const int tm = tix >> 7, tn = tix & 127;     // 4 x 128 = 512 tiles, 1/wave
      const int n = tn * 16 + ln;
      const int r = tm * 16 + ln;
      const float bias = b_ih1[n] + b_hh1[n];
      v8f c;
      #pragma unroll
      for (int i = 0; i < 8; ++i) c[i] = bias;
      const float* wi = W_ih1 + (long)n * (EE + KVD);
      const float* wh = W_hh1 + (long)n * HH;
      c = wmma_span(c, smem + A_EMB + r * 260,       256, wi);
      c = wmma_span(c, smem + A_CTX + r * 132,       128, wi + 256);
      c = wmma_span(c, smem + A_H1  + r * 516,       256, wh);
      c = wmma_span(c, smem + A_H1  + r * 516 + 258, 256, wh + 256);
      store_tile(c, gates1, 4 * HH, tm * 16, n, 4 * HH);
    }
    grid_barrier(bar, bar + 1);

    // ---- Phase 2: LSTM1 elementwise (i,f,g,o) ----
    {
      const int gsz = NWG * NTHREADS;
      for (int i = blockIdx.x * NTHREADS + tid; i < BB * HH; i += gsz) {
        const int r = i / HH, j = i - r * HH;
        const float* g = gates1 + (long)r * 4 * HH;
        const float ci = sigmf(g[HH + j]) * c1[i] + sigmf(g[j]) * tanhf(g[2 * HH + j]);
        c1[i] = ci;
        h1[i] = sigmf(g[3 * HH + j]) * tanhf(ci);
      }
    }
    grid_barrier(bar, bar + 1);

    // ---- Phase 3 stage: h1 -> A_H1, h2 -> A_CTX ----
#if HAVE_TDM
    if (wave == 0) {
      tdm_load_contig(h1, (unsigned)(A_H1 * 4),  BB * HH,  7u, 1u);
      tdm_load_contig(h2, (unsigned)(A_CTX * 4), BB * KVD, 6u, 3u);
      __builtin_amdgcn_s_wait_tensorcnt(0);
    }
#else
    for (int idx = tid; idx < BB * HH; idx += NTHREADS) {
      const int r = idx >> 9, k = idx & 511;
      smem[A_H1 + r * 516 + k + 2 * (k >= 256)] = h1[idx];
    }
    for (int idx = tid; idx < BB * KVD; idx += NTHREADS) {
      const int r = idx >> 7, k = idx & 127;
      smem[A_CTX + r * 132 + k] = h2[idx];
    }
#endif
    __syncthreads();

    // ---- Phase 3 WMMA: gates2 = [h1|h2] @ [W_ih2|W_hh2]^T + b ----
    if (tix < 128) {                               // 4 x 32 = 128 tiles
      const int tm = tix >> 5, tn = tix & 31;
      const int n = tn * 16 + ln;
      const int r = tm * 16 + ln;
      const float bias = b_ih2[n] + b_hh2[n];
      v8f c;
      #pragma unroll
      for (int i = 0; i < 8; ++i) c[i] = bias;
      const float* wi = W_ih2 + (long)n * HH;
      const float* wh = W_hh2 + (long)n * KVD;
      c = wmma_span(c, smem + A_H1  + r * 516,       256, wi);
      c = wmma_span(c, smem + A_H1  + r * 516 + 258, 256, wi + 256);
      c = wmma_span(c, smem + A_CTX + r * 132,       128, wh);
      store_tile(c, gates2, 4 * KVD, tm * 16, n, 4 * KVD);
    }
    grid_barrier(bar, bar + 1);

    // ---- Phase 4: LSTM2 elementwise ----
    {
      const int gsz = NWG * NTHREADS;
      for (int i = blockIdx.x * NTHREADS + tid; i < BB * KVD; i += gsz) {
        const int r = i / KVD, j = i - r * KVD;
        const float* g = gates2 + (long)r * 4 * KVD;
        const float ci = sigmf(g[KVD + j]) * c2[i] + sigmf(g[j]) * tanhf(g[2 * KVD + j]);
        c2[i] = ci;
        h2[i] = sigmf(g[3 * KVD + j]) * tanhf(ci);
      }
    }
    grid_barrier(bar, bar + 1);

    // ---- Phase 5: attention with new h2 -> ctx ; record attn row for b=0 ----
    attention_phase(smem, h2, enc_key, enc_value, enc_len, ctx, attns + (long)t * SS);
    grid_barrier(bar, bar + 1);

    // ---- Phase 6 stage: h2 -> A_CTX, ctx -> A_EMB (stride-132 panels) ----
#if HAVE_TDM
    if (wave == 0) {
      tdm_load_contig(h2,  (unsigned)(A_CTX * 4), BB * KVD, 6u, 3u);
      tdm_load_contig(ctx, (unsigned)(A_EMB * 4), BB * KVD, 6u, 3u);
      __builtin_amdgcn_s_wait_tensorcnt(0);
    }
#else
    for (int idx = tid; idx < BB * KVD; idx += NTHREADS) {
      const int r = idx >> 7, k = idx & 127;
      smem[A_CTX + r * 132 + k] = h2[idx];
      smem[A_EMB + r * 132 + k] = ctx[idx];
    }
#endif
    __syncthreads();

    // ---- Phase 6 WMMA: predictions[:,t,:] = [h2|ctx] @ W_emb^T + b_out ----
    if (tix < 252) {                               // 4 x ceil(1000/16) = 252 tiles
      const int tm = tix / 63, tn = tix - tm * 63;
      const int n = tn * 16 + ln;
      const int nc = (n < VV) ? n : (VV - 1);      // clamp: padded lanes stay in-bounds
      const int r = tm * 16 + ln;
      const float bias = (n < VV) ? b_out[n] : 0.0f;
      v8f c;
      #pragma unroll
      for (int i = 0; i < 8; ++i) c[i] = bias;
      const float* wr = W_emb + (long)nc * EE;
      c = wmma_span(c, smem + A_CTX + r * 132, 128, wr);
      c = wmma_span(c, smem + A_EMB + r * 132, 128, wr + 128);
      store_tile(c, preds + (long)t * VV, (long)TT * VV, tm * 16, n, VV);
    }
    __syncthreads();   // smem reused by next iteration's phase-1 staging
  }
}

extern "C" void kernel_launch(void* const* d_in, const int* in_sizes, int n_in,
                              void* d_out, int out_size, void* d_ws, size_t ws_size,
                              hipStream_t stream) {
  const float* enc_key   = (const float*)d_in[0];
  const float* enc_value = (const float*)d_in[1];
  const int*   enc_len   = (const int*)  d_in[2];
  const int*   y         = (const int*)  d_in[3];
  const float* W_emb     = (const float*)d_in[4];
  const float* b_out     = (const float*)d_in[5];
  const float* W_ih1     = (const float*)d_in[6];
  const float* W_hh1     = (const float*)d_in[7];
  const float* b_ih1     = (const float*)d_in[8];
  const float* b_hh1     = (const float*)d_in[9];
  const float* W_ih2     = (const float*)d_in[10];
  const float* W_hh2     = (const float*)d_in[11];
  const float* b_ih2     = (const float*)d_in[12];
  const float* b_hh2     = (const float*)d_in[13];

  las_init<<<NWG, NTHREADS, 0, stream>>>((float*)d_ws);
  las_decoder<<<NWG, NTHREADS, SMEM_FLOATS * sizeof(float), stream>>>(
      enc_key, enc_value, enc_len, y, W_emb, b_out,
      W_ih1, W_hh1, b_ih1, b_hh1, W_ih2, W_hh2, b_ih2, b_hh2,
      (float*)d_out, (float*)d_ws);
}